// HierarchicalSoftmaxLoss_77532749627815
// MI455X (gfx1250) — compile-verified
//
#include <hip/hip_runtime.h>
#include <hip/hip_bf16.h>
#include <math.h>

typedef __attribute__((ext_vector_type(2))) float v2f;
typedef __attribute__((ext_vector_type(8))) float v8f;

// Hierarchical softmax loss:
//   loss = (1/B) * sum_b sum_{l=0}^{L-1} softplus(-scores[b, (1<<l)-1 + bit_l])
// where bit_l = (class_indices[b] >> (L-1-l)) & 1, L = ceil(log2(V)).
//
// Single block of 1024 threads (32 waves). Each thread gathers 16 scattered
// f32 values for its row and accumulates stable softplus. Partials staged in
// LDS as a 16x64 f32 matrix; wave 0 reduces it with 16 chained
// V_WMMA_F32_16X16X4_F32 ops against an all-ones B (exact f32 sum), then a
// tiny shuffle finishes the scalar. Fully deterministic (no float atomics).
__global__ __launch_bounds__(1024)
void HierarchicalSoftmaxLoss_77532749627815_kernel(const float* __restrict__ scores,
                                                   const int* __restrict__ cls,
                                                   float* __restrict__ out,
                                                   int B, int V, int L) {
    __shared__ float p[1024];  // viewed as P[16][64], P[m][k] = p[m*64+k]

    const int tid = threadIdx.x;

    // ---- Phase 1: per-thread gather + softplus accumulation ----
    float partial = 0.0f;
    for (int b = tid; b < B; b += 1024) {
        const int ci = cls[b];
        const float* row = scores + (size_t)b * (size_t)V;
        float acc = 0.0f;
        for (int l = 0; l < L; ++l) {
            const int bit = (ci >> (L - 1 - l)) & 1;
            const int col = (1 << l) - 1 + bit;
            const float x = row[col];
            // -log(sigmoid(x)) = softplus(-x) = max(-x,0) + log1p(exp(-|x|))
            acc += fmaxf(-x, 0.0f) + log1pf(expf(-fabsf(x)));
        }
        partial += acc;
    }
    p[tid] = partial;
    __syncthreads();

    // ---- Phase 2: wave-0 reduction of 1024 partials via f32 WMMA ----
    if (tid < 32) {
        const int m    = tid & 15;   // row of 16x64 matrix handled by this lane
        const int half = tid >> 4;   // 0 -> K={0,1}, 1 -> K={2,3} within chunk

        v2f bones;
        bones.x = 1.0f; bones.y = 1.0f;   // B = 4x16 all-ones
        v8f acc8 = {};                    // C/D accumulator (16x16 f32)

        #pragma unroll
        for (int c = 0; c < 16; ++c) {    // contract K = 0..63 in chunks of 4
            const int k0 = 4 * c + 2 * half;
            v2f a;
            a.x = p[m * 64 + k0];
            a.y = p[m * 64 + k0 + 1];
            // D = A(16x4) * ones(4x16) + C : each row of D becomes rowsum(A)
            acc8 = __builtin_amdgcn_wmma_f32_16x16x4_f32(
                /*neg_a=*/false, a, /*neg_b=*/false, bones,
                /*c_mod=*/(short)0, acc8, /*reuse_a=*/false, /*reuse_b=*/false);
        }

        // Lanes 0-15: sum of rowsums M=0..7; lanes 16-31: M=8..15.
        float s = acc8[0] + acc8[1] + acc8[2] + acc8[3]
                + acc8[4] + acc8[5] + acc8[6] + acc8[7];
        const float total = __shfl(s, 0, 32) + __shfl(s, 16, 32);

        if (tid == 0) {
            out[0] = total / (float)B;
        }
    }
}

extern "C" void kernel_launch(void* const* d_in, const int* in_sizes, int n_in,
                              void* d_out, int out_size, void* d_ws, size_t ws_size,
                              hipStream_t stream) {
    const float* scores = (const float*)d_in[0];
    const int*   cls    = (const int*)d_in[1];
    float*       out    = (float*)d_out;

    const int B = in_sizes[1];
    const int V = in_sizes[0] / B;          // 50257
    int L = 0;
    while ((1LL << L) < (long long)V) ++L;  // ceil(log2(V)) = 16

    HierarchicalSoftmaxLoss_77532749627815_kernel<<<1, 1024, 0, stream>>>(
        scores, cls, out, B, V, L);
}